// FavorMaskingAttention_11716670783497
// MI455X (gfx1250) — compile-verified
//
#include <hip/hip_runtime.h>
#include <hip/hip_bf16.h>

#define B_DIM 8
#define L_DIM 32768
#define D_DIM 64
#define TOPK  64
#define EPS   1e-3f

typedef __attribute__((ext_vector_type(2))) float v2f;
typedef __attribute__((ext_vector_type(8))) float v8f;

// ---------------------------------------------------------------------------
// Workspace layout (floats):
//   [0, 512)        reduced_q[8][64]
//   [512, 520)      cutoff[8]
//   [1024, 263168)  scores[8][32768]
// ---------------------------------------------------------------------------

__global__ void k_zero_ws(float* __restrict__ ws) {
    unsigned i = threadIdx.x;
    if (i < 520) ws[i] = 0.0f;
}

// ---------------------------------------------------------------------------
// Pass 1: reduced_q[b,d] = sum_l (relu(q[b,l,d]) + EPS)
// grid (64 chunks, 8 batches), block 256. Each chunk = 512 rows.
// Thread t keeps d-group t%16 fixed (stride 256 preserves mod 16), so it can
// accumulate a private float4, then LDS ds_add_f32, then one global atomic.
// ---------------------------------------------------------------------------
__global__ void k_reduce_q(const float* __restrict__ q, float* __restrict__ rq) {
    const int b     = blockIdx.y;
    const int chunk = blockIdx.x;
    const int t     = threadIdx.x;

    const float4* qv = (const float4*)(q + (size_t)b * L_DIM * D_DIM) +
                       (size_t)chunk * 512 * 16;

    float4 acc = {0.f, 0.f, 0.f, 0.f};
#pragma unroll
    for (int i = 0; i < 32; ++i) {
        float4 v = qv[t + i * 256];
        acc.x += fmaxf(v.x, 0.f) + EPS;
        acc.y += fmaxf(v.y, 0.f) + EPS;
        acc.z += fmaxf(v.z, 0.f) + EPS;
        acc.w += fmaxf(v.w, 0.f) + EPS;
    }

    __shared__ float lds[64];
    if (t < 64) lds[t] = 0.f;
    __syncthreads();
    const int d4 = (t & 15) * 4;
    atomicAdd(&lds[d4 + 0], acc.x);
    atomicAdd(&lds[d4 + 1], acc.y);
    atomicAdd(&lds[d4 + 2], acc.z);
    atomicAdd(&lds[d4 + 3], acc.w);
    __syncthreads();
    if (t < 64) atomicAdd(&rq[b * 64 + t], lds[t]);
}

// ---------------------------------------------------------------------------
// Pass 2: scores[b,l] = rq[b,:] . (relu(k[b,l,:]) + EPS) via WMMA f32 16x16x4.
//   A (16x4, f32, 2 VGPRs): lane m   -> A[m, k0..k0+1] (vgpr0/1)
//                           lane m+16-> A[m, k0+2..k0+3]
//   B (4x16): all 16 columns = rq slice -> every column of C is the scores.
// Chain 16 WMMAs to cover D=64 for a 16-row tile. Lanes 0 and 16 hold the
// full score column (C vgpr j = M=j / M=8+j) and store 8 floats each.
// grid (64, 8), block 256 (8 waves), 4 tiles per wave -> 2048 tiles/batch.
// ---------------------------------------------------------------------------
__global__ void k_scores_wmma(const float* __restrict__ keys,
                              const float* __restrict__ rq,
                              float* __restrict__ scores) {
    const int b = blockIdx.y;

    __shared__ float rql[64];
    if (threadIdx.x < 64) rql[threadIdx.x] = rq[b * 64 + threadIdx.x];
    __syncthreads();

    const int wave = threadIdx.x >> 5;
    const int lane = threadIdx.x & 31;
    const int off  = (lane >> 4) << 1;   // 0 for lanes 0-15, 2 for 16-31
    const int row  = lane & 15;

    // Per-lane B operand slices (broadcast rq across all 16 columns).
    v2f bv[16];
#pragma unroll
    for (int kk = 0; kk < 16; ++kk) {
        bv[kk].x = rql[4 * kk + off + 0];
        bv[kk].y = rql[4 * kk + off + 1];
    }

    const float* kbase = keys + (size_t)b * L_DIM * D_DIM;

    for (int tt = 0; tt < 4; ++tt) {
        const int  tile = (blockIdx.x * 8 + wave) * 4 + tt;  // 0..2047
        const int  l0   = tile * 16;
        const float* rowp = kbase + (size_t)(l0 + row) * D_DIM + off;

        v8f c = {0.f, 0.f, 0.f, 0.f, 0.f, 0.f, 0.f, 0.f};
#pragma unroll
        for (int kk = 0; kk < 16; ++kk) {
            v2f a = *(const v2f*)(rowp + 4 * kk);   // 8B aligned (off even)
            a.x = fmaxf(a.x, 0.f) + EPS;
            a.y = fmaxf(a.y, 0.f) + EPS;
            c = __builtin_amdgcn_wmma_f32_16x16x4_f32(
                    false, a, false, bv[kk], (short)0, c, false, false);
        }

        if ((lane & 15) == 0) {
            float* sp = scores + (size_t)b * L_DIM + l0 + ((lane >> 4) << 3);
            float4 lo = {c[0], c[1], c[2], c[3]};
            float4 hi = {c[4], c[5], c[6], c[7]};
            *(float4*)(sp + 0) = lo;
            *(float4*)(sp + 4) = hi;
        }
    }
}

// ---------------------------------------------------------------------------
// Pass 3: exact radix-select of the 65th-largest score per batch.
// Monotonic uint transform; 4 byte-wise passes over 32768 scores (L2-hot).
// ---------------------------------------------------------------------------
__device__ __forceinline__ unsigned f2mono(float f) {
    unsigned u = __float_as_uint(f);
    return (u & 0x80000000u) ? ~u : (u | 0x80000000u);
}

__global__ void k_cutoff(const float* __restrict__ scores, float* __restrict__ cutoff) {
    const int b = blockIdx.x;
    const float* sp = scores + (size_t)b * L_DIM;

    __shared__ unsigned hist[256];
    __shared__ unsigned s_prefix, s_rank;
    if (threadIdx.x == 0) { s_prefix = 0u; s_rank = TOPK; }  // rank 64 desc (0-based)
    unsigned prefix_mask = 0u;
    __syncthreads();

    for (int shift = 24; shift >= 0; shift -= 8) {
        hist[threadIdx.x] = 0u;
        __syncthreads();
        const unsigned prefix = s_prefix;
        for (int i = threadIdx.x; i < L_DIM; i += 256) {
            unsigned u = f2mono(sp[i]);
            if ((u & prefix_mask) == prefix)
                atomicAdd(&hist[(u >> shift) & 255u], 1u);
        }
        __syncthreads();
        if (threadIdx.x == 0) {
            unsigned cum = 0u, rank = s_rank;
            for (int j = 255; j >= 0; --j) {
                unsigned cnt = hist[j];
                if (cum + cnt > rank) {
                    s_rank   = rank - cum;
                    s_prefix = prefix | ((unsigned)j << shift);
                    break;
                }
                cum += cnt;
            }
        }
        prefix_mask |= (255u << shift);
        __syncthreads();
    }

    if (threadIdx.x == 0) {
        unsigned u  = s_prefix;
        unsigned fb = (u & 0x80000000u) ? (u & 0x7fffffffu) : ~u;
        cutoff[b] = __uint_as_float(fb);
    }
}

// ---------------------------------------------------------------------------
// Pass 4: out[b,l,:] = scores[b,l] > cutoff[b] ? values[b,l,:] : 0
// One float4 per thread; 8*32768*16 = 4,194,304 float4s.
// ---------------------------------------------------------------------------
__global__ void k_mask(const float* __restrict__ values,
                       const float* __restrict__ scores,
                       const float* __restrict__ cutoff,
                       float* __restrict__ out) {
    const size_t idx = (size_t)blockIdx.x * 256 + threadIdx.x;
    const size_t l   = idx >> 4;                 // global row index (b*L + l)
    const int    b   = (int)(l >> 15);           // L = 32768 = 2^15
    const float  s   = scores[l];
    const float  ct  = cutoff[b];
    float4 v = ((const float4*)values)[idx];
    float4 z = {0.f, 0.f, 0.f, 0.f};
    ((float4*)out)[idx] = (s > ct) ? v : z;
}

// ---------------------------------------------------------------------------
extern "C" void kernel_launch(void* const* d_in, const int* in_sizes, int n_in,
                              void* d_out, int out_size, void* d_ws, size_t ws_size,
                              hipStream_t stream) {
    const float* q = (const float*)d_in[0];
    const float* k = (const float*)d_in[1];
    const float* v = (const float*)d_in[2];
    // d_in[3] is top_k == 64 (fixed by the reference); hardcoded as TOPK.
    (void)in_sizes; (void)n_in; (void)out_size; (void)ws_size;

    float* ws      = (float*)d_ws;
    float* rq      = ws;          // 512 floats
    float* cut     = ws + 512;    // 8 floats
    float* scores  = ws + 1024;   // 262144 floats (~1 MB)
    float* out     = (float*)d_out;

    k_zero_ws<<<1, 1024, 0, stream>>>(ws);

    dim3 g1(64, 8);
    k_reduce_q<<<g1, 256, 0, stream>>>(q, rq);

    dim3 g2(64, 8);
    k_scores_wmma<<<g2, 256, 0, stream>>>(k, rq, scores);

    k_cutoff<<<8, 256, 0, stream>>>(scores, cut);

    k_mask<<<16384, 256, 0, stream>>>(v, scores, cut, out);
}